// GroupLorentzConv2d_19121194401875
// MI455X (gfx1250) — compile-verified
//
#include <hip/hip_runtime.h>
#include <math.h>

typedef float v2f __attribute__((ext_vector_type(2)));
typedef float v8f __attribute__((ext_vector_type(8)));

#define CIN      65
#define HW       4096
#define COUT     256
#define LINF     577    // (CIN-1)*9 + 1
#define WSTRIDE  580    // padded/permuted K extent (even -> b64-aligned rows)

// ---------------------------------------------------------------------------
// Prep 1: t_resc[b, p] = sqrt( sum_{3x3} max(xpad[b,0,...],1)^2 - 8 )
// ---------------------------------------------------------------------------
__global__ void prep_tresc(const float* __restrict__ x, float* __restrict__ tws) {
    int idx = blockIdx.x * blockDim.x + threadIdx.x;   // 0 .. 16383
    int b  = idx >> 12;
    int p  = idx & 4095;
    int py = p >> 6, px = p & 63;
    const float* xb = x + b * CIN * HW;                // channel 0 plane
    float s = 0.0f;
    #pragma unroll
    for (int du = -1; du <= 1; ++du) {
        #pragma unroll
        for (int dv = -1; dv <= 1; ++dv) {
            int yy = py + du, xx = px + dv;
            float v = 0.0f;
            if ((unsigned)yy < 64u && (unsigned)xx < 64u) v = xb[yy * 64 + xx];
            float t = fmaxf(v, 1.0f);
            s += t * t;
        }
    }
    tws[idx] = sqrtf(s - 8.0f);
}

// ---------------------------------------------------------------------------
// Prep 2: repack W into WT[n][f'] with permuted K and even stride 580:
//   f' < 576  -> W[n, 1 + f'] ; f' == 576 -> W[n, 0] ; f' > 576 -> 0
// ---------------------------------------------------------------------------
__global__ void prep_repack(const float* __restrict__ Wm, float* __restrict__ wt2) {
    int idx = blockIdx.x * blockDim.x + threadIdx.x;
    if (idx >= COUT * WSTRIDE) return;
    int n = idx / WSTRIDE;
    int f = idx - n * WSTRIDE;
    float v = 0.0f;
    if (f < 576)       v = Wm[n * LINF + 1 + f];
    else if (f == 576) v = Wm[n * LINF];
    wt2[idx] = v;
}

// ---------------------------------------------------------------------------
// Main: rotated-patch-gather fused fp32 WMMA GEMM + Lorentz epilogue.
// Block = 128 threads = 4 waves (2 M x 2 N); wave = 32 rows x 128 cols.
// Inner loop software-pipelined: chunk c4+1 loads issued before chunk c4 WMMAs.
// grid = (4096/64, 1, 16), z = b*4 + g.
// ---------------------------------------------------------------------------
__global__ __launch_bounds__(128) void glconv_gemm(
    const float* __restrict__ x, const float* __restrict__ wt2,
    const float* __restrict__ bias, const float* __restrict__ tws,
    float* __restrict__ out)
{
    __shared__ float red[2][2][32];

    const int tid    = threadIdx.x;
    const int lane   = tid & 31;
    const int wave   = tid >> 5;        // 0..3
    const int wn     = wave & 1;
    const int wm     = wave >> 1;
    const int gb     = blockIdx.z;      // = b*4 + g
    const int b      = gb >> 2;
    const int g      = gb & 3;
    const int laneLo = lane & 15;
    const int laneHi = lane >> 4;

    const int rowBase = blockIdx.x * 64 + wm * 32;
    const int nBase   = wn * 128;

    const float* tb = tws + b * HW;

    // per-lane pixel coords for the two M tiles
    const int pix0 = rowBase + laneLo;
    const int pix1 = pix0 + 16;
    const int py0 = pix0 >> 6, px0 = pix0 & 63;
    const int py1 = pix1 >> 6, px1 = pix1 & 63;

    // hoisted per-lane base of the channel-1 plane (+laneHi K-pair offset)
    const float* xb1 = x + b * CIN * HW + HW + (laneHi * 2) * HW;

    // per-lane W row pointers for the 8 N tiles (laneHi selects K=+2 pair)
    const float* wb[8];
    #pragma unroll
    for (int t = 0; t < 8; ++t) {
        int n = nBase + t * 16 + laneLo;
        wb[t] = wt2 + n * WSTRIDE + laneHi * 2;
    }

    v8f acc[2][8];
    #pragma unroll
    for (int m = 0; m < 2; ++m)
        #pragma unroll
        for (int t = 0; t < 8; ++t) acc[m][t] = v8f{};

    #pragma unroll 1
    for (int k9 = 0; k9 < 9; ++k9) {
        // rotation mapping (k9 loop-uniform, g block-uniform)
        int a3 = k9 / 3, b3 = k9 - 3 * a3;
        int u, v;
        if (g == 0)      { u = a3;     v = b3;     }
        else if (g == 1) { u = b3;     v = 2 - a3; }
        else if (g == 2) { u = 2 - a3; v = 2 - b3; }
        else             { u = 2 - b3; v = a3;     }
        const int du = u - 1, dv = v - 1;

        // A-side: clamped base pointer + 0/1 mask per M tile (branch-free)
        const float* ap[2];
        float amask[2];
        {
            int yy0 = py0 + du, xx0 = px0 + dv;
            amask[0] = ((unsigned)yy0 < 64u && (unsigned)xx0 < 64u) ? 1.0f : 0.0f;
            ap[0] = xb1 + (min(max(yy0, 0), 63) * 64 + min(max(xx0, 0), 63));
            int yy1 = py1 + du, xx1 = px1 + dv;
            amask[1] = ((unsigned)yy1 < 64u && (unsigned)xx1 < 64u) ? 1.0f : 0.0f;
            ap[1] = xb1 + (min(max(yy1, 0), 63) * 64 + min(max(xx1, 0), 63));
        }
        const float* wk[8];
        #pragma unroll
        for (int t = 0; t < 8; ++t) wk[t] = wb[t] + k9 * 64;

        // -------- software-pipelined 16 chunks of K=4 --------
        v2f af[2][2];   // [parity][m]
        v2f bf[2][8];   // [parity][t]

        // prologue: load chunk 0 into parity 0
        #pragma unroll
        for (int m = 0; m < 2; ++m) {
            af[0][m].x = ap[m][0]  * amask[m];
            af[0][m].y = ap[m][HW] * amask[m];
        }
        #pragma unroll
        for (int t = 0; t < 8; ++t) bf[0][t] = *(const v2f*)(wk[t]);

        #pragma unroll
        for (int c4 = 0; c4 < 16; ++c4) {
            const int cur = c4 & 1, nxt = cur ^ 1;
            if (c4 < 15) {   // prefetch chunk c4+1
                #pragma unroll
                for (int m = 0; m < 2; ++m) {
                    af[nxt][m].x = ap[m][((c4 + 1) * 4    ) * HW] * amask[m];
                    af[nxt][m].y = ap[m][((c4 + 1) * 4 + 1) * HW] * amask[m];
                }
                #pragma unroll
                for (int t = 0; t < 8; ++t)
                    bf[nxt][t] = *(const v2f*)(wk[t] + (c4 + 1) * 4);
            }
            #pragma unroll
            for (int t = 0; t < 8; ++t) {
                acc[0][t] = __builtin_amdgcn_wmma_f32_16x16x4_f32(
                    false, af[cur][0], false, bf[cur][t], (short)0, acc[0][t], false, false);
                acc[1][t] = __builtin_amdgcn_wmma_f32_16x16x4_f32(
                    false, af[cur][1], false, bf[cur][t], (short)0, acc[1][t], false, false);
            }
        }
    }

    // final K-chunk: f' = 576 (t_resc) + zero padding 577..579
    {
        v2f af0, af1;
        af0.x = laneHi ? 0.0f : tb[pix0];  af0.y = 0.0f;
        af1.x = laneHi ? 0.0f : tb[pix1];  af1.y = 0.0f;
        v2f bf[8];
        #pragma unroll
        for (int t = 0; t < 8; ++t) bf[t] = *(const v2f*)(wb[t] + 576);
        #pragma unroll
        for (int t = 0; t < 8; ++t) {
            acc[0][t] = __builtin_amdgcn_wmma_f32_16x16x4_f32(
                false, af0, false, bf[t], (short)0, acc[0][t], false, false);
            acc[1][t] = __builtin_amdgcn_wmma_f32_16x16x4_f32(
                false, af1, false, bf[t], (short)0, acc[1][t], false, false);
        }
    }

    // ---- epilogue: bias, stores (o>=1), per-row sum of squares ----
    const int outBase = gb * HW * COUT;

    float sq[2][8];
    #pragma unroll
    for (int m = 0; m < 2; ++m)
        #pragma unroll
        for (int i = 0; i < 8; ++i) sq[m][i] = 0.0f;

    #pragma unroll
    for (int t = 0; t < 8; ++t) {
        int   n   = nBase + t * 16 + laneLo;
        float bv  = bias[n];
        bool  ch0 = (n == 0);          // only laneLo==0, t==0, wn==0
        #pragma unroll
        for (int m = 0; m < 2; ++m) {
            int rBase = rowBase + m * 16 + laneHi * 8;  // lanes>=16 hold M=i+8
            #pragma unroll
            for (int i = 0; i < 8; ++i) {
                float y = acc[m][t][i] + bv;
                if (!ch0) {
                    sq[m][i] += y * y;
                    out[outBase + (rBase + i) * COUT + n] = y;
                }
            }
        }
    }

    // reduce each sq over the 16 lanes of each half (xor bits 0..3)
    #pragma unroll
    for (int m = 0; m < 2; ++m)
        #pragma unroll
        for (int i = 0; i < 8; ++i) {
            float s = sq[m][i];
            s += __shfl_xor(s, 1, 32);
            s += __shfl_xor(s, 2, 32);
            s += __shfl_xor(s, 4, 32);
            s += __shfl_xor(s, 8, 32);
            if (laneLo == 0)           // lane 0 -> rows i, lane 16 -> rows i+8
                red[wm][wn][m * 16 + laneHi * 8 + i] = s;
        }
    __syncthreads();

    // channel 0 = sqrt(sum_{o>=1} y^2 + 1); one lane per row
    if (wn == 0) {
        float tot = red[wm][0][lane] + red[wm][1][lane] + 1.0f;
        int row = rowBase + lane;      // 32 rows per wm half
        out[outBase + row * COUT] = sqrtf(tot);
    }
}

// ---------------------------------------------------------------------------
extern "C" void kernel_launch(void* const* d_in, const int* in_sizes, int n_in,
                              void* d_out, int out_size, void* d_ws, size_t ws_size,
                              hipStream_t stream) {
    const float* x    = (const float*)d_in[0];   // (4, 65, 64, 64) fp32 (flat)
    const float* Wm   = (const float*)d_in[1];   // (256, 577) fp32
    const float* bias = (const float*)d_in[2];   // (256,) fp32
    float*       out  = (float*)d_out;           // (4, 4, 64, 64, 256) fp32

    float* tws = (float*)d_ws;                   // 16384 floats = 64 KB
    float* wt2 = tws + 16384;                    // 256*580 floats ~ 594 KB (8B aligned)

    prep_tresc <<<dim3(16384 / 256),                256, 0, stream>>>(x, tws);
    prep_repack<<<dim3((COUT * WSTRIDE + 255)/256), 256, 0, stream>>>(Wm, wt2);
    glconv_gemm<<<dim3(64, 1, 16),                  128, 0, stream>>>(x, wt2, bias, tws, out);
}